// JEPAModel_28845000360381
// MI455X (gfx1250) — compile-verified
//
#include <hip/hip_runtime.h>
#include <hip/hip_bf16.h>
#include <math.h>

typedef __attribute__((ext_vector_type(2))) float v2f;
typedef __attribute__((ext_vector_type(8))) float v8f;
typedef __attribute__((ext_vector_type(4))) unsigned int v4u;
typedef __attribute__((ext_vector_type(8))) int v8i;
typedef __attribute__((ext_vector_type(4))) int v4i;

#define OBS_DIM    128
#define ACT_DIM    32
#define LATENT_DIM 512
#define BELIEF_DIM 1024
#define GATES      (3 * BELIEF_DIM)          // 3072
#define XDIM       (LATENT_DIM + ACT_DIM)    // 544

#define SCAN_WGS     32
#define SCAN_THREADS 384                      // 96 rows * 4 lanes, 12 waves

#if defined(__AMDGCN__) && __has_builtin(__builtin_amdgcn_tensor_load_to_lds) && \
    __has_builtin(__builtin_amdgcn_s_wait_tensorcnt)
#define USE_TDM 1
#else
#define USE_TDM 0
#endif

__device__ __forceinline__ float sigmoidf_(float x) {
  return 1.0f / (1.0f + __expf(-x));
}

#if USE_TDM
// Minimal 1-D Tensor-DMA descriptor (CDNA5 ISA):
// group0: count=1 | lds_addr[63:32] | global_addr[120:64] | type=2 [127:126]
// group1: data_size=2 (4B) [17:16] | tensor_dim0 [79:48] | tensor_dim1=1
//         [111:80] | tile_dim0 [127:112] | tensor_dim0_stride [207:160]
__device__ __forceinline__ void tdm_load_1d_f32(unsigned lds_off,
                                                const float* gsrc,
                                                unsigned nelem) {
  const unsigned long long ga = (unsigned long long)(uintptr_t)gsrc;
  v4u g0;
  g0[0] = 1u;                                   // count=1, user descriptor
  g0[1] = lds_off;                              // lds_addr (bytes)
  g0[2] = (unsigned)(ga & 0xffffffffu);         // global_addr[31:0]
  g0[3] = (unsigned)((ga >> 32) & 0x1ffffffu)   // global_addr[56:32]
          | (2u << 30);                         // type=2 ("image")
  v8i g1;
  g1[0] = (int)(2u << 16);                          // data_size = 4 bytes
  g1[1] = (int)((nelem & 0xffffu) << 16);           // tensor_dim0[15:0]
  g1[2] = (int)(((nelem >> 16) & 0xffffu)           // tensor_dim0[31:16]
                | (1u << 16));                      // tensor_dim1 = 1
  g1[3] = (int)((nelem & 0xffffu) << 16);           // tile_dim0 (<=65535)
  g1[4] = 0;                                        // tile_dim1/2 unused
  g1[5] = (int)nelem;                               // tensor_dim0_stride lo
  g1[6] = 0;
  g1[7] = 0;
  const v4i zz4 = {0, 0, 0, 0};                     // groups 2/3 unused
  const v8i zz8 = {0, 0, 0, 0, 0, 0, 0, 0};
  // 6-arg toolchain variant (clang-23 / therock-10.0 headers).
  __builtin_amdgcn_tensor_load_to_lds(g0, g1, zz4, zz4, zz8, 0);
}
#endif

// ---------------------------------------------------------------------------
// Init: zero the grid-barrier counter and the h double-buffer (h0 = 0).
// ---------------------------------------------------------------------------
__global__ void jepa_init_kernel(unsigned* __restrict__ counter,
                                 float* __restrict__ hbuf) {
  if (threadIdx.x == 0) *counter = 0u;
  for (int i = threadIdx.x; i < 2 * BELIEF_DIM; i += blockDim.x) hbuf[i] = 0.0f;
}

// ---------------------------------------------------------------------------
// f32 WMMA 16x16x4 fragment convention:
// A frag: lane L, vgpr j -> A[m0 + (L&15)][k + j + 2*(L>>4)]   (float2 load)
// B frag: lane L, vgpr j -> W[n0 + (L&15)][k + j + 2*(L>>4)]   (float2 load)
// D:      lane L, vgpr r -> C[m0 + r + 8*(L>>4)][n0 + (L&15)]
// Each wave computes a 32x64 tile (2 M-subtiles x 4 N-subtiles), with
// software-pipelined fragment loads (prefetch k+4 before issuing k's WMMAs).
// ---------------------------------------------------------------------------
#define WMMA_F32(A, B, C) \
  __builtin_amdgcn_wmma_f32_16x16x4_f32(false, (A), false, (B), (short)0, (C), \
                                        false, false)

// Encoder GEMM: z[m,n] = tanh(sum_k obs[m,k] * enc_w[n,k] + enc_b[n])
__global__ __launch_bounds__(256) void jepa_enc_kernel(
    const float* __restrict__ obs, const float* __restrict__ encw,
    const float* __restrict__ encb, float* __restrict__ z, int T) {
  const int wave = threadIdx.x >> 5;
  const int lane = threadIdx.x & 31;
  const int lo = lane & 15, hi = lane >> 4;
  const int NG = LATENT_DIM / 64;  // 8 n-groups
  const int gwt = blockIdx.x * 8 + wave;
  const int m0 = (gwt / NG) * 32;
  const int n0 = (gwt % NG) * 64;
  if (m0 >= T) return;

  v8f acc[2][4] = {};
  const float* a0p = obs + (size_t)(m0 + lo) * OBS_DIM + 2 * hi;
  const float* a1p = a0p + (size_t)16 * OBS_DIM;
  const float* bp[4];
#pragma unroll
  for (int nt = 0; nt < 4; ++nt)
    bp[nt] = encw + (size_t)(n0 + nt * 16 + lo) * OBS_DIM + 2 * hi;

  v2f a0 = *(const v2f*)a0p;
  v2f a1 = *(const v2f*)a1p;
  v2f b[4];
#pragma unroll
  for (int nt = 0; nt < 4; ++nt) b[nt] = *(const v2f*)bp[nt];

#pragma unroll 4
  for (int k = 0; k < OBS_DIM; k += 4) {
    const int kn = (k + 4 < OBS_DIM) ? k + 4 : 0;  // harmless tail reload
    v2f a0n = *(const v2f*)(a0p + kn);
    v2f a1n = *(const v2f*)(a1p + kn);
    v2f bn[4];
#pragma unroll
    for (int nt = 0; nt < 4; ++nt) bn[nt] = *(const v2f*)(bp[nt] + kn);
#pragma unroll
    for (int nt = 0; nt < 4; ++nt) {
      acc[0][nt] = WMMA_F32(a0, b[nt], acc[0][nt]);
      acc[1][nt] = WMMA_F32(a1, b[nt], acc[1][nt]);
    }
    a0 = a0n;
    a1 = a1n;
#pragma unroll
    for (int nt = 0; nt < 4; ++nt) b[nt] = bn[nt];
  }

#pragma unroll
  for (int mt = 0; mt < 2; ++mt) {
#pragma unroll
    for (int nt = 0; nt < 4; ++nt) {
      const int n = n0 + nt * 16 + lo;
      const float bias = encb[n];
#pragma unroll
      for (int r = 0; r < 8; ++r) {
        const int m = m0 + mt * 16 + r + 8 * hi;
        z[(size_t)m * LATENT_DIM + n] = tanhf(acc[mt][nt][r] + bias);
      }
    }
  }
}

// Gate-input GEMM: gi[m,n] = sum_k z[m,k]*w_ih[n,k]
//                          + sum_k2 a_prev[m,k2]*w_ih[n,512+k2] + b_ih[n]
__global__ __launch_bounds__(256) void jepa_gih_kernel(
    const float* __restrict__ z, const float* __restrict__ act,
    const float* __restrict__ wih, const float* __restrict__ bih,
    float* __restrict__ gi, int T) {
  const int wave = threadIdx.x >> 5;
  const int lane = threadIdx.x & 31;
  const int lo = lane & 15, hi = lane >> 4;
  const int NG = GATES / 64;  // 48 n-groups
  const int gwt = blockIdx.x * 8 + wave;
  const int m0 = (gwt / NG) * 32;
  const int n0 = (gwt % NG) * 64;
  if (m0 >= T) return;

  v8f acc[2][4] = {};
  const float* a0p = z + (size_t)(m0 + lo) * LATENT_DIM + 2 * hi;
  const float* a1p = a0p + (size_t)16 * LATENT_DIM;
  const float* bp[4];
#pragma unroll
  for (int nt = 0; nt < 4; ++nt)
    bp[nt] = wih + (size_t)(n0 + nt * 16 + lo) * XDIM + 2 * hi;

  v2f a0 = *(const v2f*)a0p;
  v2f a1 = *(const v2f*)a1p;
  v2f b[4];
#pragma unroll
  for (int nt = 0; nt < 4; ++nt) b[nt] = *(const v2f*)bp[nt];

#pragma unroll 2
  for (int k = 0; k < LATENT_DIM; k += 4) {
    const int kn = (k + 4 < LATENT_DIM) ? k + 4 : 0;
    v2f a0n = *(const v2f*)(a0p + kn);
    v2f a1n = *(const v2f*)(a1p + kn);
    v2f bn[4];
#pragma unroll
    for (int nt = 0; nt < 4; ++nt) bn[nt] = *(const v2f*)(bp[nt] + kn);
#pragma unroll
    for (int nt = 0; nt < 4; ++nt) {
      acc[0][nt] = WMMA_F32(a0, b[nt], acc[0][nt]);
      acc[1][nt] = WMMA_F32(a1, b[nt], acc[1][nt]);
    }
    a0 = a0n;
    a1 = a1n;
#pragma unroll
    for (int nt = 0; nt < 4; ++nt) b[nt] = bn[nt];
  }

  // Action part: k in [512, 544). A row m uses act[m-1], zeros at m==0.
#pragma unroll
  for (int mt = 0; mt < 2; ++mt) {
    const int m = m0 + mt * 16 + lo;
    const float* actrow = act + (size_t)(m - 1) * ACT_DIM + 2 * hi;
#pragma unroll
    for (int k = 0; k < ACT_DIM; k += 4) {
      v2f a;
      if (m > 0) a = *(const v2f*)(actrow + k);
      else { a[0] = 0.0f; a[1] = 0.0f; }
#pragma unroll
      for (int nt = 0; nt < 4; ++nt) {
        v2f bb = *(const v2f*)(bp[nt] + LATENT_DIM + k);
        acc[mt][nt] = WMMA_F32(a, bb, acc[mt][nt]);
      }
    }
  }

#pragma unroll
  for (int mt = 0; mt < 2; ++mt) {
#pragma unroll
    for (int nt = 0; nt < 4; ++nt) {
      const int n = n0 + nt * 16 + lo;
      const float bias = bih[n];
#pragma unroll
      for (int r = 0; r < 8; ++r) {
        const int m = m0 + mt * 16 + r + 8 * hi;
        gi[(size_t)m * GATES + n] = acc[mt][nt][r] + bias;
      }
    }
  }
}

// ---------------------------------------------------------------------------
// Persistent GRU scan. 32 WGs; WG g owns h-indices [g*32, g*32+32) and rows
// {j, 1024+j, 2048+j} of w_hh, so all gates for its slice are local.
// h(t) is staged into LDS via the Tensor Data Mover (1-D 4KB tile) each step.
// One monotonic grid barrier per step; h double-buffered in global memory.
// ---------------------------------------------------------------------------
__global__ __launch_bounds__(SCAN_THREADS) void jepa_scan_kernel(
    const float* __restrict__ gi, const float* __restrict__ whh,
    const float* __restrict__ bhh, float* __restrict__ out,
    float* __restrict__ hbuf, unsigned* __restrict__ counter, int T) {
  __shared__ float hs[BELIEF_DIM];
  __shared__ float dots[96];

  const int g = blockIdx.x;
  const int tid = threadIdx.x;
  const int lr = tid >> 2;     // local row 0..95
  const int q = tid & 3;       // k-slice within row
  const int gate = lr >> 5;    // 0=r, 1=z, 2=n
  const int jj = lr & 31;
  const int grow = gate * BELIEF_DIM + g * 32 + jj;
  const float4* __restrict__ wrow4 =
      (const float4*)(whh + (size_t)grow * BELIEF_DIM);
  const float bias = bhh[grow];
#if USE_TDM
  const unsigned hs_lds_off = (unsigned)(uintptr_t)(void*)hs;  // addr[31:0]
#endif

  for (int t = 0; t < T; ++t) {
    // Stage h(t) into LDS (previous step's writes ordered by grid barrier).
#if USE_TDM
    if (tid < 32) {  // one TDM issue from wave 0 (EXEC ignored by TDM)
      tdm_load_1d_f32(hs_lds_off, hbuf + (t & 1) * BELIEF_DIM, BELIEF_DIM);
      __builtin_amdgcn_s_wait_tensorcnt(0);
    }
#else
    const float4* hb4 = (const float4*)(hbuf + (t & 1) * BELIEF_DIM);
    if (tid < BELIEF_DIM / 4) ((float4*)hs)[tid] = hb4[tid];
#endif
    __syncthreads();

    // Partial dot: lane q covers k = 4q + 16i.
    float s = 0.0f;
    const float4* h4 = (const float4*)hs;
#pragma unroll 8
    for (int i = 0; i < BELIEF_DIM / 16; ++i) {
      const float4 wv = wrow4[q + 4 * i];
      const float4 hv = h4[q + 4 * i];
      s = fmaf(wv.x, hv.x, s);
      s = fmaf(wv.y, hv.y, s);
      s = fmaf(wv.z, hv.z, s);
      s = fmaf(wv.w, hv.w, s);
    }
    s += __shfl_xor(s, 1);
    s += __shfl_xor(s, 2);
    if (q == 0) dots[lr] = s + bias;  // gh + b_hh for this row
    __syncthreads();

    if (tid < 32) {
      const int j = g * 32 + tid;
      const float* git = gi + (size_t)t * GATES;
      const float r  = sigmoidf_(git[j] + dots[tid]);
      const float zg = sigmoidf_(git[BELIEF_DIM + j] + dots[32 + tid]);
      const float n  = tanhf(git[2 * BELIEF_DIM + j] + r * dots[64 + tid]);
      const float hnew = (1.0f - zg) * n + zg * hs[j];
      hbuf[((t + 1) & 1) * BELIEF_DIM + j] = hnew;
      out[(size_t)t * BELIEF_DIM + j] = hnew;
      __threadfence();  // release h_new before barrier arrive
      if (t + 1 < T) __builtin_prefetch(git + GATES + j, 0, 1);
    }
    __syncthreads();

    // Monotonic grid barrier (no reset -> no reuse race).
    if (tid == 0) {
      __hip_atomic_fetch_add(counter, 1u, __ATOMIC_RELEASE,
                             __HIP_MEMORY_SCOPE_AGENT);
      const unsigned target = (unsigned)(t + 1) * SCAN_WGS;
      while (__hip_atomic_load(counter, __ATOMIC_ACQUIRE,
                               __HIP_MEMORY_SCOPE_AGENT) < target) {
        __builtin_amdgcn_s_sleep(2);
      }
    }
    __syncthreads();
  }
}

// ---------------------------------------------------------------------------
extern "C" void kernel_launch(void* const* d_in, const int* in_sizes, int n_in,
                              void* d_out, int out_size, void* d_ws,
                              size_t ws_size, hipStream_t stream) {
  const float* obs  = (const float*)d_in[0];
  const float* act  = (const float*)d_in[1];
  const float* encw = (const float*)d_in[2];
  const float* encb = (const float*)d_in[3];
  const float* wih  = (const float*)d_in[4];
  const float* whh  = (const float*)d_in[5];
  const float* bih  = (const float*)d_in[6];
  const float* bhh  = (const float*)d_in[7];
  float* out = (float*)d_out;
  const int T = in_sizes[0] / OBS_DIM;  // 32768

  char* ws = (char*)d_ws;
  unsigned* counter = (unsigned*)ws;                         // [0, 256)
  float* hbuf = (float*)(ws + 256);                          // 2*1024 f32
  float* zbuf = (float*)(ws + 16384);                        // T*512 f32
  float* gibuf =
      (float*)(ws + 16384 + (size_t)T * LATENT_DIM * sizeof(float));  // T*3072

  jepa_init_kernel<<<1, 256, 0, stream>>>(counter, hbuf);

  const int enc_blocks = (T / 32) * (LATENT_DIM / 64) / 8;   // 1024
  jepa_enc_kernel<<<enc_blocks, 256, 0, stream>>>(obs, encw, encb, zbuf, T);

  const int gih_blocks = (T / 32) * (GATES / 64) / 8;        // 6144
  jepa_gih_kernel<<<gih_blocks, 256, 0, stream>>>(zbuf, act, wih, bih, gibuf, T);

  jepa_scan_kernel<<<SCAN_WGS, SCAN_THREADS, 0, stream>>>(gibuf, whh, bhh, out,
                                                          hbuf, counter, T);
}